// MOLELinear_6493990551579
// MI455X (gfx1250) — compile-verified
//
#include <hip/hip_runtime.h>

typedef __attribute__((ext_vector_type(2))) float v2f;
typedef __attribute__((ext_vector_type(8))) float v8f;

#define NUM_EXPERTS 16
#define FEAT 64
#define MATSZ (FEAT * FEAT) /* 4096 */

// ---------------------------------------------------------------------------
// Kernel 1: mix expert weights per system: Wm[s] = sum_e coef[s,e] * W[e]
// 32 systems x 64x64 f32 = 512 KB -> lives in d_ws, L2-resident afterwards.
// ---------------------------------------------------------------------------
__global__ void __launch_bounds__(256)
mole_mix_weights(const float* __restrict__ W,
                 const float* __restrict__ coef,
                 float* __restrict__ Wm) {
  const int s = blockIdx.x;
  float c[NUM_EXPERTS];
#pragma unroll
  for (int e = 0; e < NUM_EXPERTS; ++e) c[e] = coef[s * NUM_EXPERTS + e];
  for (int idx = threadIdx.x; idx < MATSZ; idx += blockDim.x) {
    float acc = 0.f;
#pragma unroll
    for (int e = 0; e < NUM_EXPERTS; ++e) acc += c[e] * W[e * MATSZ + idx];
    Wm[s * MATSZ + idx] = acc;
  }
}

// ---------------------------------------------------------------------------
// Kernel 2: one wave32 per 16-atom tile.
// D[16x64] = X_tile[16x64] @ Wm[s]^T  via v_wmma_f32_16x16x4_f32
//   A (16x4 f32): lane l (l<16) holds A[l][0],A[l][1]; lane l+16 holds A[l][2],A[l][3]
//   B (4x16 f32): lane l holds B[0][l],B[1][l]; lane l+16 holds B[2][l],B[3][l]
//   C/D (16x16 f32): vgpr j, lanes 0-15 -> row j, lanes 16-31 -> row j+8
// batch_indices sorted => tile spans systems [s_lo, s_hi]; loop over them
// (wave-uniform scalar loop; EXEC stays all-1s as WMMA requires). Row results
// are committed with an exact masked-FMA (res += (rowSys==s) * acc), which
// lowers to v_cndmask + v_fmac instead of saveexec branches.
// ---------------------------------------------------------------------------
__global__ void __launch_bounds__(256)
mole_main(const float* __restrict__ x,
          const int* __restrict__ bidx,
          const float* __restrict__ Wm,
          const float* __restrict__ bias,
          float* __restrict__ out,
          int N) {
  const int lane = threadIdx.x & 31;
  const int wave = threadIdx.x >> 5;
  const int tile = blockIdx.x * (blockDim.x >> 5) + wave;
  const int base = tile * 16;
  if (base >= N) return;  // wave-uniform

  const int half = lane >> 4;   // 0: K lo pair / rows 0-7 ; 1: K hi pair / rows 8-15
  const int l    = lane & 15;

  // Systems spanned by this tile (indices are sorted).
  const int hiAtom = (base + 15 < N) ? base + 15 : N - 1;
  const int s_lo = __builtin_amdgcn_readfirstlane(bidx[base]);
  const int s_hi = __builtin_amdgcn_readfirstlane(bidx[hiAtom]);

  // Preload all 16 A-fragments (the whole 16x64 x-tile, 4 KB) once.
  int arow = base + l;
  if (arow >= N) arow = N - 1;
  const float* xrow = x + (size_t)arow * FEAT + 2 * half;
  v2f afrag[16];
#pragma unroll
  for (int k16 = 0; k16 < 16; ++k16)
    afrag[k16] = *(const v2f*)(xrow + 4 * k16);

  // System id of the output row each accumulator element j maps to.
  int rowSys[8];
#pragma unroll
  for (int j = 0; j < 8; ++j) {
    int atom = base + j + 8 * half;
    rowSys[j] = bidx[(atom < N) ? atom : N - 1];
  }

  v8f res[4] = {{}, {}, {}, {}};

  for (int s = s_lo; s <= s_hi; ++s) {
    const float* Wb = Wm + s * MATSZ + 2 * half;
#pragma unroll
    for (int nt = 0; nt < 4; ++nt) {
      const float* Wc = Wb + (nt * 16 + l) * FEAT;  // row o = nt*16 + l of Wm[s]
      v8f acc = {};
#pragma unroll
      for (int k16 = 0; k16 < 16; ++k16) {
        v2f b = *(const v2f*)(Wc + 4 * k16);
        acc = __builtin_amdgcn_wmma_f32_16x16x4_f32(
            /*neg_a=*/false, afrag[k16], /*neg_b=*/false, b,
            /*c_mod=*/(short)0, acc, /*reuse_a=*/false, /*reuse_b=*/false);
      }
      // Exact masked commit: each row's system matches exactly one s, and
      // res starts at 0, so res == acc for that s. v_cndmask + v_fmac,
      // no EXEC-mask branches.
#pragma unroll
      for (int j = 0; j < 8; ++j) {
        const float m = (rowSys[j] == s) ? 1.0f : 0.0f;
        res[nt][j] = __builtin_fmaf(m, acc[j], res[nt][j]);
      }
    }
  }

  // Bias add + store (lanes 0-15 cover cols nt*16..nt*16+15 of rows j / j+8).
#pragma unroll
  for (int nt = 0; nt < 4; ++nt) {
    const float bv = bias[nt * 16 + l];
#pragma unroll
    for (int j = 0; j < 8; ++j) {
      const int atom = base + j + 8 * half;
      if (atom < N)
        out[(size_t)atom * FEAT + nt * 16 + l] = res[nt][j] + bv;
    }
  }
}

// ---------------------------------------------------------------------------
extern "C" void kernel_launch(void* const* d_in, const int* in_sizes, int n_in,
                              void* d_out, int out_size, void* d_ws, size_t ws_size,
                              hipStream_t stream) {
  const float* x    = (const float*)d_in[0];  // [N, 64]
  const float* coef = (const float*)d_in[1];  // [S, 16]
  const int*   bidx = (const int*)d_in[2];    // [N] sorted
  const float* W    = (const float*)d_in[3];  // [16, 64, 64]
  const float* bias = (const float*)d_in[4];  // [64]
  float* out = (float*)d_out;

  const int N = in_sizes[2];
  const int S = in_sizes[1] / NUM_EXPERTS;

  float* Wm = (float*)d_ws;  // S * 4096 floats = 512 KB scratch

  mole_mix_weights<<<S, 256, 0, stream>>>(W, coef, Wm);

  const int tiles  = (N + 15) / 16;
  const int wavesB = 256 / 32;
  const int blocks = (tiles + wavesB - 1) / wavesB;
  mole_main<<<blocks, 256, 0, stream>>>(x, bidx, Wm, bias, out, N);
}